// SampleKSoftmaxUnbiasedWithTrimmedLassoGate_3856880632037
// MI455X (gfx1250) — compile-verified
//
#include <hip/hip_runtime.h>
#include <hip/hip_bf16.h>

typedef float v2f __attribute__((ext_vector_type(2)));
typedef float v4f __attribute__((ext_vector_type(4)));
typedef float v8f __attribute__((ext_vector_type(8)));

#define DIM   768
#define NEXP  8
#define KTOP  2
#define TAU_  1.0f

// ---------------------------------------------------------------------------
// Kernel 0: zero the 16 accumulator floats (soft[8], hard[8]) in workspace.
// ---------------------------------------------------------------------------
__global__ void init_sums_kernel(float* sums) {
    if (threadIdx.x < 16) sums[threadIdx.x] = 0.0f;
}

// ---------------------------------------------------------------------------
// Kernel 1: gate logits via V_WMMA_F32_16X16X4_F32 + softmax/top2 gating.
// Block = 256 threads = 8 waves; each wave owns a 16-row tile -> 128 rows/blk.
// ---------------------------------------------------------------------------
__global__ __launch_bounds__(256) void gate_kernel(
    const float* __restrict__ x,      // [B, 768]
    const float* __restrict__ W,      // [8, 768]
    const float* __restrict__ bias,   // [8]
    float* __restrict__ g_out,        // [B, 8]  (workspace)
    float* __restrict__ sums)         // [16]    (workspace atomics)
{
    __shared__ float lds_logits[8 * 16 * NEXP];  // 8 waves x 16 rows x 8 experts
    __shared__ float lds_soft[NEXP];
    __shared__ float lds_hard[NEXP];

    const int tid  = threadIdx.x;
    const int wave = tid >> 5;
    const int lane = tid & 31;
    const int row0 = blockIdx.x * 128 + wave * 16;

    if (tid < NEXP) { lds_soft[tid] = 0.0f; lds_hard[tid] = 0.0f; }

    // A-matrix 16x4 f32 layout: lane = m + 16*khalf, VGPR0/1 = K = 2*khalf + {0,1}
    const int m     = lane & 15;
    const int khalf = lane >> 4;
    const float* xrow = x + (size_t)(row0 + m) * DIM + 2 * khalf;
    // B-matrix 4x16 (W^T): lane = n + 16*khalf; pad experts 8..15 with zeros.
    const float  wscale = (m < NEXP) ? 1.0f : 0.0f;
    const float* wrow   = W + (size_t)(m & (NEXP - 1)) * DIM + 2 * khalf;

    v8f acc = {0.f, 0.f, 0.f, 0.f, 0.f, 0.f, 0.f, 0.f};

    #pragma unroll 8
    for (int k0 = 0; k0 < DIM; k0 += 4) {
        v2f a = *(const v2f*)(xrow + k0);
        v2f bm = *(const v2f*)(wrow + k0);
        bm.x *= wscale;
        bm.y *= wscale;
        // D = A(16x4) * B(4x16) + C, f32 accumulate
        acc = __builtin_amdgcn_wmma_f32_16x16x4_f32(
            /*neg_a=*/false, a, /*neg_b=*/false, bm,
            /*c_mod=*/(short)0, acc, /*reuse_a=*/false, /*reuse_b=*/false);
    }

    // D layout: lane L holds D[m][n], n = L&15, m = vgprIdx + 8*(L>>4)
    float* myTile = lds_logits + wave * (16 * NEXP);
    {
        const int nn  = lane & 15;
        const int mhi = (lane >> 4) * 8;
        if (nn < NEXP) {
            #pragma unroll
            for (int i = 0; i < 8; ++i)
                myTile[(mhi + i) * NEXP + nn] = acc[i];
        }
    }
    __syncthreads();

    // Lanes 0..15 each own one row: softmax / top-2 / adjusted sparse softmax.
    if (lane < 16) {
        const int row = row0 + lane;
        float lg[NEXP];
        #pragma unroll
        for (int e = 0; e < NEXP; ++e)
            lg[e] = (myTile[lane * NEXP + e] + bias[e]) * (1.0f / TAU_);

        // top-1 (ties -> lower index, matching lax.top_k)
        float m1 = lg[0]; int i1 = 0;
        #pragma unroll
        for (int e = 1; e < NEXP; ++e)
            if (lg[e] > m1) { m1 = lg[e]; i1 = e; }
        // top-2
        float m2 = -INFINITY; int i2 = 0;
        #pragma unroll
        for (int e = 0; e < NEXP; ++e)
            if (e != i1 && lg[e] > m2) { m2 = lg[e]; i2 = e; }

        // dense softmax g
        float ex[NEXP], s = 0.0f;
        #pragma unroll
        for (int e = 0; e < NEXP; ++e) { ex[e] = __expf(lg[e] - m1); s += ex[e]; }
        const float inv = 1.0f / s;

        // adj = logits - log(clip(K*(g+1e-10), 1e-10)), only needed at i1,i2
        const float g1 = ex[i1] * inv, g2 = ex[i2] * inv;
        const float a1 = lg[i1] - __logf(fmaxf((float)KTOP * (g1 + 1e-10f), 1e-10f));
        const float a2 = lg[i2] - __logf(fmaxf((float)KTOP * (g2 + 1e-10f), 1e-10f));
        const float am = fmaxf(a1, a2);
        const float e1 = __expf(a1 - am), e2 = __expf(a2 - am);
        const float rs = 1.0f / (e1 + e2);
        const float gs1 = e1 * rs, gs2 = e2 * rs;

        float ov[NEXP];
        #pragma unroll
        for (int e = 0; e < NEXP; ++e) {
            const float v = (e == i1) ? gs1 : ((e == i2) ? gs2 : 0.0f);
            ov[e] = v;
            atomicAdd(&lds_soft[e], v);
            atomicAdd(&lds_hard[e], (v >= 1e-5f) ? 1.0f : 0.0f);
        }
        v4f* gp = (v4f*)(g_out + (size_t)row * NEXP);
        v4f o0 = {ov[0], ov[1], ov[2], ov[3]};
        v4f o1 = {ov[4], ov[5], ov[6], ov[7]};
        gp[0] = o0;
        gp[1] = o1;
    }
    __syncthreads();

    if (tid < NEXP) {
        atomicAdd(&sums[tid],        lds_soft[tid]);
        atomicAdd(&sums[NEXP + tid], lds_hard[tid]);
    }
}

// ---------------------------------------------------------------------------
// Kernel 2: y[b,d] = sum_e h[b,d,e] * g[b,e]  — pure HBM streaming (192 MiB).
// One block per batch row; NT loads/stores (zero reuse, h ~= whole L2).
// ---------------------------------------------------------------------------
__global__ __launch_bounds__(256) void combine_kernel(
    const float* __restrict__ h,   // [B, 768, 8]
    const float* __restrict__ g,   // [B, 8]
    float* __restrict__ y)         // [B, 768]
{
    const int b = blockIdx.x;
    const v4f* gv = (const v4f*)(g + (size_t)b * NEXP);
    const v4f g0 = gv[0];
    const v4f g1 = gv[1];
    const float* hb = h + (size_t)b * DIM * NEXP;
    float*       yb = y + (size_t)b * DIM;

    #pragma unroll
    for (int it = 0; it < DIM / 256; ++it) {
        const int i = it * 256 + threadIdx.x;
        const v4f* hp = (const v4f*)(hb + (size_t)i * NEXP);
        const v4f h0 = __builtin_nontemporal_load(hp);
        const v4f h1 = __builtin_nontemporal_load(hp + 1);
        float acc = h0.x * g0.x;
        acc = fmaf(h0.y, g0.y, acc);
        acc = fmaf(h0.z, g0.z, acc);
        acc = fmaf(h0.w, g0.w, acc);
        acc = fmaf(h1.x, g1.x, acc);
        acc = fmaf(h1.y, g1.y, acc);
        acc = fmaf(h1.z, g1.z, acc);
        acc = fmaf(h1.w, g1.w, acc);
        __builtin_nontemporal_store(acc, yb + i);
    }
}

// ---------------------------------------------------------------------------
// Kernel 3: averages = sums / BATCH into the output tail (soft[8], hard[8]).
// ---------------------------------------------------------------------------
__global__ void finalize_kernel(const float* sums, float* out_tail, float inv_batch) {
    if (threadIdx.x < 16) out_tail[threadIdx.x] = sums[threadIdx.x] * inv_batch;
}

// ---------------------------------------------------------------------------
extern "C" void kernel_launch(void* const* d_in, const int* in_sizes, int n_in,
                              void* d_out, int out_size, void* d_ws, size_t ws_size,
                              hipStream_t stream) {
    const float* h    = (const float*)d_in[0];   // [B, 768, 8]
    const float* x    = (const float*)d_in[1];   // [B, 768]
    const float* W    = (const float*)d_in[2];   // [8, 768]
    const float* bias = (const float*)d_in[3];   // [8]
    float* out = (float*)d_out;

    const int batch = in_sizes[1] / DIM;         // 8192

    float* sums = (float*)d_ws;                  // 16 floats
    float* g    = sums + 16;                     // batch * 8 floats

    init_sums_kernel<<<1, 32, 0, stream>>>(sums);
    gate_kernel<<<batch / 128, 256, 0, stream>>>(x, W, bias, g, sums);
    combine_kernel<<<batch, 256, 0, stream>>>(h, g, out);
    finalize_kernel<<<1, 32, 0, stream>>>(sums, out + (size_t)batch * DIM,
                                          1.0f / (float)batch);
}